// AuthPctMetric_82652350644508
// MI455X (gfx1250) — compile-verified
//
#include <hip/hip_runtime.h>

// AuthPct metric for MI455X (gfx1250, wave32).
// Two fused 12288x12288x256 f32 WMMA GEMM-distance kernels with
// double-buffered GLOBAL_LOAD_ASYNC_TO_LDS staging (ASYNCcnt) and
// in-epilogue column min / argmin reduction via integer atomics.

typedef float v2f __attribute__((ext_vector_type(2)));
typedef float v8f __attribute__((ext_vector_type(8)));

#define N_PTS 12288
#define KDIM  256
#define BLK   128      // output tile edge per workgroup
#define KC    32       // K chunk staged in LDS
#define NCHUNK (KDIM / KC)
#define LDT   36       // padded LDS row stride (floats): 144B, 16B-aligned, bank-spread
#define BIGSQ 1.0e20f  // (1e10)^2 diagonal fill in squared-distance domain

// ---------------------------------------------------------------------------
// Kernel 1: row norms for real & gen + init of the min-reduction arrays.
// ---------------------------------------------------------------------------
__global__ __launch_bounds__(256) void norms_init_kernel(
    const float* __restrict__ real, const float* __restrict__ gen,
    float* __restrict__ r2, float* __restrict__ g2,
    unsigned int* __restrict__ minr, unsigned long long* __restrict__ minkey) {
  const int i = blockIdx.x * blockDim.x + threadIdx.x;
  if (i >= N_PTS) return;
  const float4* rp = (const float4*)&real[(size_t)i * KDIM];
  const float4* gp = (const float4*)&gen[(size_t)i * KDIM];
  float sr = 0.f, sg = 0.f;
#pragma unroll 4
  for (int q = 0; q < KDIM / 4; ++q) {
    float4 a = rp[q];
    sr += a.x * a.x + a.y * a.y + a.z * a.z + a.w * a.w;
    float4 b = gp[q];
    sg += b.x * b.x + b.y * b.y + b.z * b.z + b.w * b.w;
  }
  r2[i] = sr;
  g2[i] = sg;
  minr[i] = 0x7F7FFFFFu;                     // FLT_MAX bit pattern
  minkey[i] = 0xFFFFFFFFFFFFFFFFull;
}

// ---------------------------------------------------------------------------
// Async global->LDS staging of one 128x32 f32 tile pair (A rows, B rows).
// GVS addressing: SGPR 64-bit base + per-lane 32-bit byte offset.
// Per thread: 4 + 4 B128 async transfers, tracked on ASYNCcnt.
// ---------------------------------------------------------------------------
__device__ __forceinline__ void stage_async_chunk(
    const float* __restrict__ A, const float* __restrict__ B, int row0,
    int col0, int tid, int kb, unsigned ldsA0, unsigned ldsB0) {
#pragma unroll
  for (int it = 0; it < 4; ++it) {
    const int f = tid + it * 256;  // float4 index within 128x32 tile
    const int r = f >> 3;          // row within tile
    const int kq = f & 7;          // float4 within K-chunk
    const unsigned goffA = (unsigned)(((row0 + r) * KDIM + kb + kq * 4) * 4);
    const unsigned goffB = (unsigned)(((col0 + r) * KDIM + kb + kq * 4) * 4);
    const unsigned la = ldsA0 + (unsigned)((r * LDT + kq * 4) * 4);
    const unsigned lb = ldsB0 + (unsigned)((r * LDT + kq * 4) * 4);
    asm volatile("global_load_async_to_lds_b128 %0, %1, %2"
                 :
                 : "v"(la), "v"(goffA), "s"(A)
                 : "memory");
    asm volatile("global_load_async_to_lds_b128 %0, %1, %2"
                 :
                 : "v"(lb), "v"(goffB), "s"(B)
                 : "memory");
  }
}

// ---------------------------------------------------------------------------
// Kernel 2: tiled WMMA f32 GEMM (dot products) fused with squared-distance
// epilogue and per-column min (IS_REAL) or min+argmin (gen) reduction.
// Block = 256 threads = 8 waves (2 along M x 4 along N). Wave tile 64x32.
// ---------------------------------------------------------------------------
template <int IS_REAL>
__global__ __launch_bounds__(256) void dist_min_kernel(
    const float* __restrict__ A,   // rows i: real [N,K]
    const float* __restrict__ B,   // cols j: real or gen [N,K]
    const float* __restrict__ r2,  // |A_i|^2
    const float* __restrict__ c2,  // |B_j|^2
    unsigned int* __restrict__ minr,
    unsigned long long* __restrict__ minkey) {
  __shared__ float As[2][BLK * LDT];
  __shared__ float Bs[2][BLK * LDT];
  __shared__ float sR2[BLK];
  __shared__ float sC2[BLK];

  const int tid = threadIdx.x;
  const int row0 = blockIdx.y * BLK;
  const int col0 = blockIdx.x * BLK;

  if (tid < BLK)
    sR2[tid] = r2[row0 + tid];
  else
    sC2[tid - BLK] = c2[col0 + tid - BLK];

  const int w = tid >> 5;        // wave id 0..7
  const int lane = tid & 31;
  const int waveM = w >> 2;      // 0..1 -> 64-row slab
  const int waveN = w & 3;       // 0..3 -> 32-col slab
  const int lane_n = lane & 15;  // M (A frag) / N (B frag, C cols)
  const int hi = lane >> 4;      // half-wave: K offset +2 for frags, M +8 for C

  const unsigned asb = (unsigned)(unsigned long long)&As[0][0];
  const unsigned bsb = (unsigned)(unsigned long long)&Bs[0][0];
  const unsigned bufsz = (unsigned)(BLK * LDT * 4);

  v8f acc[4][2];
#pragma unroll
  for (int s = 0; s < 4; ++s)
#pragma unroll
    for (int t = 0; t < 2; ++t)
#pragma unroll
      for (int e = 0; e < 8; ++e) acc[s][t][e] = 0.0f;

  // Prologue: async-fill buffer 0 with chunk 0.
  stage_async_chunk(A, B, row0, col0, tid, 0, asb, bsb);
  asm volatile("s_wait_asynccnt 0x0" ::: "memory");
  __syncthreads();  // also covers sR2/sC2

  for (int c = 0; c < NCHUNK; ++c) {
    const int cur = c & 1;
    if (c + 1 < NCHUNK) {  // overlap next chunk's fill with this chunk's WMMAs
      const unsigned nb = (unsigned)((c + 1) & 1) * bufsz;
      stage_async_chunk(A, B, row0, col0, tid, (c + 1) * KC, asb + nb, bsb + nb);
    }
    const float* __restrict__ Asc = &As[cur][0];
    const float* __restrict__ Bsc = &Bs[cur][0];

#pragma unroll
    for (int kk = 0; kk < KC; kk += 4) {
      const int kf = kk + 2 * hi;  // ISA f32 16x16x4 layout: half-wave K split
      v2f a[4], b[2];
#pragma unroll
      for (int s = 0; s < 4; ++s)
        a[s] = *(const v2f*)&Asc[(waveM * 64 + s * 16 + lane_n) * LDT + kf];
#pragma unroll
      for (int t = 0; t < 2; ++t)
        b[t] = *(const v2f*)&Bsc[(waveN * 32 + t * 16 + lane_n) * LDT + kf];
#pragma unroll
      for (int s = 0; s < 4; ++s)
#pragma unroll
        for (int t = 0; t < 2; ++t)
          acc[s][t] = __builtin_amdgcn_wmma_f32_16x16x4_f32(
              false, a[s], false, b[t], (short)0, acc[s][t], false, false);
    }

    // Own async fill complete + all waves done reading buf `cur` before the
    // next iteration can refill it.
    asm volatile("s_wait_asynccnt 0x0" ::: "memory");
    __syncthreads();
  }

  // Epilogue: squared distance + column min/argmin.
  // C layout: lane holds N = lane_n, M = e + 8*hi within each 16x16 subtile.
#pragma unroll
  for (int t = 0; t < 2; ++t) {
    const int jloc = waveN * 32 + t * 16 + lane_n;
    const int j = col0 + jloc;
    const float cn = sC2[jloc];
    float bestv = 3.0e38f;
    unsigned int besti = 0;
#pragma unroll
    for (int s = 0; s < 4; ++s) {
#pragma unroll
      for (int e = 0; e < 8; ++e) {
        const int iloc = waveM * 64 + s * 16 + e + 8 * hi;
        const int i = row0 + iloc;
        float sq = sR2[iloc] + cn - 2.0f * acc[s][t][e];
        if (IS_REAL && i == j) sq = BIGSQ;  // diagonal fill
        sq = fmaxf(sq, 0.0f);               // clamp -> bit pattern is monotonic
        if (sq < bestv) {
          bestv = sq;
          besti = (unsigned int)i;
        }
      }
    }
    if (IS_REAL) {
      const float o = __shfl_xor(bestv, 16, 32);  // pair halves share column j
      bestv = fminf(bestv, o);
      if (hi == 0) atomicMin(&minr[j], __float_as_uint(bestv));
    } else {
      unsigned long long key =
          ((unsigned long long)__float_as_uint(bestv) << 32) | (unsigned long long)besti;
      const unsigned long long ok = __shfl_xor(key, 16, 32);
      key = (ok < key) ? ok : key;  // min value, ties -> smallest index (argmin)
      if (hi == 0) atomicMin(&minkey[j], key);
    }
  }
}

// ---------------------------------------------------------------------------
// Kernel 3: gather + sigmoid + block partial sums.
// ---------------------------------------------------------------------------
__global__ __launch_bounds__(256) void authen_partial_kernel(
    const unsigned long long* __restrict__ minkey,
    const unsigned int* __restrict__ minr, float* __restrict__ partial) {
  __shared__ float red[256];
  const int j = blockIdx.x * blockDim.x + threadIdx.x;
  float v = 0.0f;
  if (j < N_PTS) {
    const unsigned long long key = minkey[j];
    const float d1 = sqrtf(__uint_as_float((unsigned int)(key >> 32)));
    const unsigned int idx = (unsigned int)key;
    const float d2 = sqrtf(__uint_as_float(minr[idx]));
    const float z = (d2 - d1) * 10.0f;  // /TEMP, TEMP=0.1
    v = 1.0f / (1.0f + __expf(-z));
  }
  red[threadIdx.x] = v;
  __syncthreads();
#pragma unroll
  for (int s = 128; s > 0; s >>= 1) {
    if (threadIdx.x < s) red[threadIdx.x] += red[threadIdx.x + s];
    __syncthreads();
  }
  if (threadIdx.x == 0) partial[blockIdx.x] = red[0];
}

__global__ __launch_bounds__(64) void final_reduce_kernel(
    const float* __restrict__ partial, int nparts, float* __restrict__ out) {
  __shared__ float red[64];
  float v = (threadIdx.x < nparts) ? partial[threadIdx.x] : 0.0f;
  red[threadIdx.x] = v;
  __syncthreads();
#pragma unroll
  for (int s = 32; s > 0; s >>= 1) {
    if (threadIdx.x < s) red[threadIdx.x] += red[threadIdx.x + s];
    __syncthreads();
  }
  if (threadIdx.x == 0) out[0] = -100.0f * red[0] / (float)N_PTS;
}

// ---------------------------------------------------------------------------
extern "C" void kernel_launch(void* const* d_in, const int* in_sizes, int n_in,
                              void* d_out, int out_size, void* d_ws, size_t ws_size,
                              hipStream_t stream) {
  const float* real = (const float*)d_in[0];
  const float* gen = (const float*)d_in[1];
  float* out = (float*)d_out;

  // Workspace layout (u64 array first for 8B alignment).
  unsigned long long* minkey = (unsigned long long*)d_ws;        // N_PTS u64
  float* r2 = (float*)(minkey + N_PTS);                          // N_PTS f32
  float* g2 = r2 + N_PTS;                                        // N_PTS f32
  unsigned int* minr = (unsigned int*)(g2 + N_PTS);              // N_PTS u32
  float* partial = (float*)(minr + N_PTS);                       // 48 f32

  const int nblk = N_PTS / 256;  // 48

  norms_init_kernel<<<nblk, 256, 0, stream>>>(real, gen, r2, g2, minr, minkey);

  dim3 grid(N_PTS / BLK, N_PTS / BLK);  // 96 x 96
  dist_min_kernel<1><<<grid, 256, 0, stream>>>(real, real, r2, r2, minr, minkey);
  dist_min_kernel<0><<<grid, 256, 0, stream>>>(real, gen, r2, g2, minr, minkey);

  authen_partial_kernel<<<nblk, 256, 0, stream>>>(minkey, minr, partial);
  final_reduce_kernel<<<1, 64, 0, stream>>>(partial, nblk, out);
}